// TwoStreamMovingAvgPool_9131100471429
// MI455X (gfx1250) — compile-verified
//
#include <hip/hip_runtime.h>
#include <hip/hip_bf16.h>

// Problem constants (reference: B=64, T=512, D=768, decay=0.9)
#define BB 64
#define TT 512
#define DD 768
#define DECAYF 0.9f
// log2(0.9)
#define LOG2_DECAY (-0.15200309344504997f)
// coeff threshold: 0.1 * 0.9^S < 1e-10 is numerically irrelevant vs O(1) f32 result
#define COEFF_EPS 1e-10f

typedef __attribute__((ext_vector_type(2))) float v2f;
typedef __attribute__((ext_vector_type(8))) float v8f;

// force a wave-uniform float into an SGPR (value is uniform by construction)
__device__ __forceinline__ float uniform_f32(float v) {
    return __builtin_bit_cast(float,
        __builtin_amdgcn_readfirstlane(__builtin_bit_cast(int, v)));
}

// ---------------------------------------------------------------------------
// Kernel 1: per (stream, batch) compute EMA-as-weighted-sum coefficients.
//   coeff[b,t] = m_t * (1-decay) * decay^(# active steps after t)
// Also: has_any flag and tstart = first t with coeff > eps (prefix skip).
// Block = 512 threads (one per t), grid = (B, 2).
// ---------------------------------------------------------------------------
__global__ void __launch_bounds__(512)
coeff_kernel(const float* __restrict__ spk_mask,
             const float* __restrict__ act_mask,
             float* __restrict__ ws_coeff,   // [2][B][T]
             int*   __restrict__ ws_tstart,  // [2][B]
             float* __restrict__ ws_flag)    // [2][B]
{
    const int b = blockIdx.x;
    const int s = blockIdx.y;
    const float* mask = (s ? act_mask : spk_mask) + (size_t)b * TT;

    const int t    = threadIdx.x;          // 0..511
    const int lane = threadIdx.x & 31;     // wave32
    const int wid  = threadIdx.x >> 5;     // 16 waves

    const float m = mask[t];

    // wave-level inclusive scan (wave32)
    float v = m;
    #pragma unroll
    for (int off = 1; off < 32; off <<= 1) {
        float x = __shfl_up(v, off, 32);
        if (lane >= off) v += x;
    }

    __shared__ float wsum[16];
    __shared__ float carry[16];
    if (lane == 31) wsum[wid] = v;
    __syncthreads();
    if (wid == 0) {
        float x = (lane < 16) ? wsum[lane] : 0.0f;
        #pragma unroll
        for (int off = 1; off < 16; off <<= 1) {
            float y = __shfl_up(x, off, 32);
            if (lane >= off) x += y;
        }
        if (lane < 16) carry[lane] = x;
    }
    __syncthreads();

    const float total = carry[15];
    const float incl  = v + (wid ? carry[wid - 1] : 0.0f);
    const float S     = total - incl;            // # active steps strictly after t
    const float one_minus = 1.0f - DECAYF;       // matches reference's 1.0 - decay
    const float coeff = m * one_minus * exp2f(S * LOG2_DECAY);

    ws_coeff[(((size_t)s * BB) + b) * TT + t] = coeff;

    __shared__ int tmin;
    if (threadIdx.x == 0) tmin = TT;
    __syncthreads();
    if (coeff > COEFF_EPS) atomicMin(&tmin, t);  // LDS atomic, order-independent
    __syncthreads();
    if (threadIdx.x == 0) {
        ws_tstart[s * BB + b] = tmin;
        ws_flag[s * BB + b]   = (total > 0.0f) ? 1.0f : 0.0f;
    }
}

// ---------------------------------------------------------------------------
// Kernel 2: weighted time-reduction via V_WMMA_F32_16X16X4_F32.
//   C(16x16) += A(16x4) x B(4x16), K = time chunk of 4, N = 16 D-columns,
//   A row M=0 holds the coefficients (rows 1..15 zero) -> C row 0 is the answer.
// Per ISA 7.12.2 (32-bit A 16x4): lanes 0-15 hold {K0,K1}, lanes 16-31 {K2,K3};
// B 4x16 mirrors this with N = lane%16. C f32: VGPR0 lanes 0-15 = (M=0, N=lane).
// Grid = (6, B, 2); block = 256 = 8 waves; each wave owns one 16-wide D tile.
// Coefficients are forced into SGPRs (readfirstlane), so A placement is pure
// v_cndmask from SGPRs, EXEC stays all-ones through the loop (WMMA rule), and
// the B loads strength-reduce to one pointer bump + immediate offsets.
// ---------------------------------------------------------------------------
__global__ void __launch_bounds__(256)
wmma_reduce(const float* __restrict__ spk_hist,
            const float* __restrict__ act_hist,
            const float* __restrict__ spk_mean,
            const float* __restrict__ act_mean,
            const float* __restrict__ ws_coeff,
            const int*   __restrict__ ws_tstart,
            const float* __restrict__ ws_flag,
            float* __restrict__ out)           // full d_out
{
    const int s = blockIdx.z;
    const int b = blockIdx.y;
    const int wave = threadIdx.x >> 5;
    const int l    = threadIdx.x & 31;
    const int n    = l & 15;                 // D column within tile / N index
    const int kb   = (l >> 4) << 1;          // lanes 0-15 -> K{0,1}, 16-31 -> K{2,3}
    const int d0   = blockIdx.x * 128 + wave * 16;

    const float* X    = (s ? act_hist : spk_hist) + (size_t)b * TT * DD;
    const float* mean = (s ? act_mean : spk_mean) + (size_t)b * DD;
    const float* w    = ws_coeff + (((size_t)s * BB) + b) * TT;

    // chunk-of-8 aligned prefix skip; force into SGPR (it is block-uniform)
    const int t0 = __builtin_amdgcn_readfirstlane(ws_tstart[s * BB + b] & ~7);
    const float hasAny = ws_flag[s * BB + b];

    const bool amRow0 = (n == 0);            // only row M=0 of A is non-zero
    const bool hiHalf = (kb != 0);           // lanes 16-31

    // per-lane streaming base: column d0+n, row t0+kb
    const float* q  = X + ((size_t)(t0 + kb)) * DD + d0 + n;
    const float* wt = w + t0;

    v8f acc0 = {};
    v8f acc1 = {};

    for (int t = t0; t < TT; t += 8) {
        // wave-uniform chunk coefficients, pinned to SGPRs
        const float c0 = uniform_f32(wt[0]);
        const float c1 = uniform_f32(wt[1]);
        const float c2 = uniform_f32(wt[2]);
        const float c3 = uniform_f32(wt[3]);
        const float c4 = uniform_f32(wt[4]);
        const float c5 = uniform_f32(wt[5]);
        const float c6 = uniform_f32(wt[6]);
        const float c7 = uniform_f32(wt[7]);

        v2f a0, a1;                          // branchless A placement (cndmask)
        a0[0] = amRow0 ? (hiHalf ? c2 : c0) : 0.0f;
        a0[1] = amRow0 ? (hiHalf ? c3 : c1) : 0.0f;
        a1[0] = amRow0 ? (hiHalf ? c6 : c4) : 0.0f;
        a1[1] = amRow0 ? (hiHalf ? c7 : c5) : 0.0f;

        v2f b0, b1;                          // B[k][n] = X[t+k][d0+n], streamed NT
        b0[0] = __builtin_nontemporal_load(q);
        b0[1] = __builtin_nontemporal_load(q + DD);
        b1[0] = __builtin_nontemporal_load(q + 4 * DD);
        b1[1] = __builtin_nontemporal_load(q + 5 * DD);

        acc0 = __builtin_amdgcn_wmma_f32_16x16x4_f32(
                   false, a0, false, b0, (short)0, acc0, false, false);
        acc1 = __builtin_amdgcn_wmma_f32_16x16x4_f32(
                   false, a1, false, b1, (short)0, acc1, false, false);

        q  += 8 * DD;                        // single 64-bit bump per iteration
        wt += 8;
    }

    const float r = acc0[0] + acc1[0];       // C row M=0 lives in VGPR0, lanes 0-15

    if (l < 16) {
        const float res = (hasAny > 0.0f) ? r : mean[d0 + l];
        // layout of d_out: [c | c_spk | c_act | w]; stream s -> slot (1+s)
        const size_t N = (size_t)BB * DD;
        out[(size_t)(1 + s) * N + (size_t)b * DD + d0 + l] = res;
    }
}

// ---------------------------------------------------------------------------
// Kernel 3: softmax mix  c = w0*c_spk + w1*c_act, plus the w output.
// ---------------------------------------------------------------------------
__global__ void __launch_bounds__(256)
mix_kernel(const float* __restrict__ logits, float* __restrict__ out)
{
    const float l0 = logits[0], l1 = logits[1];
    const float mx = fmaxf(l0, l1);
    const float e0 = expf(l0 - mx), e1 = expf(l1 - mx);
    const float inv = 1.0f / (e0 + e1);
    const float w0 = e0 * inv, w1 = e1 * inv;

    const int N = BB * DD;
    const int i = blockIdx.x * blockDim.x + threadIdx.x;
    if (i < N) {
        out[i] = w0 * out[N + i] + w1 * out[2 * N + i];
    }
    if (i == 0) {
        out[3 * N]     = w0;
        out[3 * N + 1] = w1;
    }
}

// ---------------------------------------------------------------------------
extern "C" void kernel_launch(void* const* d_in, const int* in_sizes, int n_in,
                              void* d_out, int out_size, void* d_ws, size_t ws_size,
                              hipStream_t stream) {
    const float* spk_hist   = (const float*)d_in[0];
    const float* spk_mask   = (const float*)d_in[1];
    const float* act_hist   = (const float*)d_in[2];
    const float* act_mask   = (const float*)d_in[3];
    const float* spk_mean   = (const float*)d_in[4];
    const float* act_mean   = (const float*)d_in[5];
    const float* mix_logits = (const float*)d_in[6];
    float* out = (float*)d_out;

    // workspace layout
    float* ws_coeff  = (float*)d_ws;                                   // 2*B*T floats
    int*   ws_tstart = (int*)((char*)d_ws + (size_t)2 * BB * TT * 4);  // 2*B ints
    float* ws_flag   = (float*)((char*)ws_tstart + (size_t)2 * BB * 4);// 2*B floats

    coeff_kernel<<<dim3(BB, 2), 512, 0, stream>>>(
        spk_mask, act_mask, ws_coeff, ws_tstart, ws_flag);

    wmma_reduce<<<dim3(DD / 128, BB, 2), 256, 0, stream>>>(
        spk_hist, act_hist, spk_mean, act_mean,
        ws_coeff, ws_tstart, ws_flag, out);

    const int N = BB * DD;
    mix_kernel<<<dim3((N + 255) / 256), 256, 0, stream>>>(mix_logits, out);
}